// QuixerCore_30339648979607
// MI455X (gfx1250) — compile-verified
//
#include <hip/hip_runtime.h>
#include <hip/hip_bf16.h>
#include <math.h>

// QuixerCore for gfx1250 (MI455X, wave32).
// Pipeline:
//   1) angles GEMM [1024x512]x[512x48] via V_WMMA_F32_16X16X4_F32 (full f32 WMMA)
//   2) prep: normalize lcu weights, init mono=basis0, acc=c0*basis0
//   3) per QSVT iter k=1..3:
//        norm_kernel   : degenerate-norm guard (replace mono by basis0 if ||mono||<EPS)
//        evolve_kernel : 1024 blocks (1 per token), state in LDS, 48 gates, scale by lcu[t]
//        reduce_kernel : deterministic sum over tokens -> mono ; acc += c_k * mono
//   4) final_kernel: acc normalization, ff ansatz, X/Y/Z expvals -> 36 f32
//
// Workspace layout (floats): angles 49152 | wre 1024 | wim 1024 | mono 8192 |
//                            acc 8192 | evolved 1024*8192  (~33.9 MB total)

#define NQ   12
#define DIM  4096
#define PAIR 2048
#define CPAIR 1024
#define NPAR 48
#define LTOK 1024
#define DMODEL 512
#define EPSV 1e-8f

typedef __attribute__((ext_vector_type(2))) float v2f;
typedef __attribute__((ext_vector_type(8))) float v8f;

// ---------------------------------------------------------------------------
// 1) Angles GEMM: C[t][p] = sum_d A[t][d] * W[p][d] + b[p]
//    One wave32 per 16x16 output tile; K=512 in steps of 4 via f32 WMMA.
//    A frag (16x4, 32-bit): lanes 0-15 M=lane (K=0/1 in v0/v1), lanes 16-31 M=lane-16 (K=2/3)
//    B frag (4x16, 32-bit): v0: K=0 (lanes 0-15) / K=2 (lanes 16-31), v1: K=1 / K=3, N=lane%16
// ---------------------------------------------------------------------------
__global__ void __launch_bounds__(32) angles_wmma_kernel(
    const float* __restrict__ A,   // [1024][512] tokens_emb
    const float* __restrict__ W,   // [48][512]   W_ang
    const float* __restrict__ b,   // [48]        b_ang
    float* __restrict__ angles)    // [1024][48]
{
    const int tile = blockIdx.x;           // 0..191
    const int tm = tile / 3, tn = tile % 3;
    const int m0 = tm * 16, n0 = tn * 16;
    const int lane = threadIdx.x;
    const int lr = lane & 15;
    const int hi = lane >> 4;              // 0: K pair {0,1}, 1: K pair {2,3}

    const float* __restrict__ arow = A + (size_t)(m0 + lr) * DMODEL;
    const float* __restrict__ wrow = W + (size_t)(n0 + lr) * DMODEL;

    v8f c = {};
    for (int k = 0; k < DMODEL; k += 4) {
        v2f av = *reinterpret_cast<const v2f*>(arow + k + 2 * hi);
        v2f bv = *reinterpret_cast<const v2f*>(wrow + k + 2 * hi);
        c = __builtin_amdgcn_wmma_f32_16x16x4_f32(
            /*neg_a=*/false, av, /*neg_b=*/false, bv,
            /*c_mod=*/(short)0, c, /*reuse_a=*/false, /*reuse_b=*/false);
    }

    const float bias = b[n0 + lr];
#pragma unroll
    for (int r = 0; r < 8; ++r) {
        const int m = m0 + r + 8 * hi;     // C layout: vgpr r -> rows r / r+8
        angles[(size_t)m * NPAR + n0 + lr] = c[r] + bias;
    }
}

// ---------------------------------------------------------------------------
// Gate primitives on an LDS-resident state (sre/sim, 4096 floats each).
// Wire w corresponds to index bit (11 - w) (reshape (2,)*12, axis 0 = MSB).
// ---------------------------------------------------------------------------
__device__ __forceinline__ void apply_ry(float* sre, float* sim, int bp,
                                         float c, float s, int tid) {
    const int m = 1 << bp;
#pragma unroll
    for (int it = 0; it < PAIR / 256; ++it) {
        const int p  = tid + it * 256;
        const int i0 = ((p & ~(m - 1)) << 1) | (p & (m - 1));
        const int i1 = i0 | m;
        const float a0r = sre[i0], a0i = sim[i0];
        const float a1r = sre[i1], a1i = sim[i1];
        sre[i0] = c * a0r - s * a1r;  sim[i0] = c * a0i - s * a1i;
        sre[i1] = s * a0r + c * a1r;  sim[i1] = s * a0i + c * a1i;
    }
}

__device__ __forceinline__ void apply_crx(float* sre, float* sim, int pc, int pt,
                                          float c, float s, int tid) {
    const int bc = 1 << pc, bt = 1 << pt;
    const int p1 = pc < pt ? pc : pt;
    const int p2 = pc < pt ? pt : pc;
#pragma unroll
    for (int it = 0; it < CPAIR / 256; ++it) {
        int x = tid + it * 256;            // 10 free bits
        x = ((x & ~((1 << p1) - 1)) << 1) | (x & ((1 << p1) - 1));
        x = ((x & ~((1 << p2) - 1)) << 1) | (x & ((1 << p2) - 1));
        const int i10 = x | bc;            // ctrl=1, tgt=0
        const int i11 = i10 | bt;          // ctrl=1, tgt=1
        const float r0 = sre[i10], q0 = sim[i10];
        const float r1 = sre[i11], q1 = sim[i11];
        // a10' = c*a10 - i*s*a11 ; a11' = -i*s*a10 + c*a11
        sre[i10] =  c * r0 + s * q1;  sim[i10] =  c * q0 - s * r1;
        sre[i11] =  s * q0 + c * r1;  sim[i11] = -s * r0 + c * q1;
    }
}

// Sim et al. ansatz 14 (1 layer), matching the reference's gate order exactly.
__device__ void run_ansatz(float* sre, float* sim,
                           const float* ch, const float* sh, int tid) {
    for (int i = 0; i < NQ; ++i) {                       // RY sweep 1
        apply_ry(sre, sim, NQ - 1 - i, ch[i], sh[i], tid);
        __syncthreads();
    }
    for (int k = 0; k < NQ; ++k) {                       // CRX ring (descending)
        const int i = NQ - 1 - k;
        const int t = (i + 1) % NQ;
        apply_crx(sre, sim, NQ - 1 - i, NQ - 1 - t, ch[NQ + k], sh[NQ + k], tid);
        __syncthreads();
    }
    for (int i = 0; i < NQ; ++i) {                       // RY sweep 2
        apply_ry(sre, sim, NQ - 1 - i, ch[2 * NQ + i], sh[2 * NQ + i], tid);
        __syncthreads();
    }
    for (int k = 0; k < NQ; ++k) {                       // CRX ring ([11,0,1..10])
        const int i = (k == 0) ? (NQ - 1) : (k - 1);
        const int t = (i - 1 + NQ) % NQ;
        apply_crx(sre, sim, NQ - 1 - i, NQ - 1 - t, ch[3 * NQ + k], sh[3 * NQ + k], tid);
        __syncthreads();
    }
}

// ---------------------------------------------------------------------------
// 2) prep: lcu normalization + weight table; mono = basis0; acc = c0 * basis0
// ---------------------------------------------------------------------------
__global__ void __launch_bounds__(1024) prep_kernel(
    const float* __restrict__ lre, const float* __restrict__ lim,
    const float* __restrict__ coeffs,
    float* __restrict__ wre, float* __restrict__ wim,
    float* __restrict__ mono, float* __restrict__ acc)
{
    __shared__ float red[1024];
    const int tid = threadIdx.x;
    const float r = lre[tid], i = lim[tid];
    red[tid] = sqrtf(r * r + i * i);
    __syncthreads();
    for (int s = 512; s > 0; s >>= 1) {
        if (tid < s) red[tid] += red[tid + s];
        __syncthreads();
    }
    const float S = fmaxf(red[0], EPSV);
    wre[tid] = r / S;
    wim[tid] = i / S;
    for (int d = tid; d < DIM; d += 1024) {
        const float m0 = (d == 0) ? 1.0f : 0.0f;
        mono[2 * d] = m0;            mono[2 * d + 1] = 0.0f;
        acc[2 * d]  = coeffs[0] * m0; acc[2 * d + 1]  = 0.0f;
    }
}

// ---------------------------------------------------------------------------
// 3a) norm guard: if ||mono|| < EPS, mono <- basis0 (linearity cancels
//     the normalize/denormalize pair in the non-degenerate branch).
// ---------------------------------------------------------------------------
__global__ void __launch_bounds__(256) norm_kernel(float* __restrict__ mono)
{
    __shared__ float red[256];
    const int tid = threadIdx.x;
    float a = 0.0f;
    for (int d = tid; d < DIM; d += 256) {
        const float r = mono[2 * d], i = mono[2 * d + 1];
        a += r * r + i * i;
    }
    red[tid] = a;
    __syncthreads();
    for (int s = 128; s > 0; s >>= 1) {
        if (tid < s) red[tid] += red[tid + s];
        __syncthreads();
    }
    if (sqrtf(red[0]) < EPSV) {
        for (int d = tid; d < DIM; d += 256) {
            mono[2 * d] = (d == 0) ? 1.0f : 0.0f;
            mono[2 * d + 1] = 0.0f;
        }
    }
}

// ---------------------------------------------------------------------------
// 3b) evolve: one block per token. State in LDS; 48 gates; write lcu[t]*U_t|mono>.
// ---------------------------------------------------------------------------
__global__ void __launch_bounds__(256) evolve_kernel(
    const float* __restrict__ mono, const float* __restrict__ angles,
    const float* __restrict__ wre, const float* __restrict__ wim,
    float* __restrict__ evolved)
{
    __shared__ float sre[DIM];
    __shared__ float sim_[DIM];
    __shared__ float ch[NPAR];
    __shared__ float sh[NPAR];
    const int t = blockIdx.x;
    const int tid = threadIdx.x;

    if (tid < NPAR) {
        const float h = 0.5f * angles[(size_t)t * NPAR + tid];
        ch[tid] = cosf(h);
        sh[tid] = sinf(h);
    }
    for (int d = tid; d < DIM; d += 256) {
        sre[d]  = mono[2 * d];
        sim_[d] = mono[2 * d + 1];
    }
    __syncthreads();

    run_ansatz(sre, sim_, ch, sh, tid);

    const float wr = wre[t], wi = wim[t];
    float* __restrict__ dst = evolved + (size_t)t * (2 * DIM);
    for (int d = tid; d < DIM; d += 256) {
        const float r = sre[d], i = sim_[d];
        dst[2 * d]     = wr * r - wi * i;
        dst[2 * d + 1] = wr * i + wi * r;
    }
}

// ---------------------------------------------------------------------------
// 3c) reduce over tokens (deterministic): mono = sum_t ; acc += c_k * mono
// ---------------------------------------------------------------------------
__global__ void __launch_bounds__(256) reduce_kernel(
    const float* __restrict__ evolved, const float* __restrict__ coeffs, int k,
    float* __restrict__ mono, float* __restrict__ acc)
{
    const int d = blockIdx.x * 256 + threadIdx.x;   // 0..4095
    float sr = 0.0f, si = 0.0f;
    for (int t = 0; t < LTOK; ++t) {
        const float* __restrict__ p = evolved + (size_t)t * (2 * DIM) + 2 * d;
        sr += p[0];
        si += p[1];
    }
    mono[2 * d] = sr;
    mono[2 * d + 1] = si;
    const float c = coeffs[k];
    acc[2 * d]     += c * sr;
    acc[2 * d + 1] += c * si;
}

// ---------------------------------------------------------------------------
// 4) final: scale acc, normalize, ff ansatz, measure X/Y/Z on all 12 wires.
// ---------------------------------------------------------------------------
__global__ void __launch_bounds__(256) final_kernel(
    const float* __restrict__ acc, const float* __restrict__ coeffs,
    const float* __restrict__ ff, float* __restrict__ out)
{
    __shared__ float sre[DIM];
    __shared__ float sim_[DIM];
    __shared__ float ch[NPAR];
    __shared__ float sh[NPAR];
    __shared__ float red[256];
    const int tid = threadIdx.x;

    const float csum = fmaxf(fabsf(coeffs[0]) + fabsf(coeffs[1]) +
                             fabsf(coeffs[2]) + fabsf(coeffs[3]), EPSV);
    if (tid < NPAR) {
        const float h = 0.5f * ff[tid];
        ch[tid] = cosf(h);
        sh[tid] = sinf(h);
    }
    float nrm = 0.0f;
    for (int d = tid; d < DIM; d += 256) {
        const float r = acc[2 * d] / csum;
        const float i = acc[2 * d + 1] / csum;
        sre[d] = r; sim_[d] = i;
        nrm += r * r + i * i;
    }
    red[tid] = nrm;
    __syncthreads();
    for (int s = 128; s > 0; s >>= 1) {
        if (tid < s) red[tid] += red[tid + s];
        __syncthreads();
    }
    const float sn = sqrtf(red[0]);
    if (sn < EPSV) {
        for (int d = tid; d < DIM; d += 256) {
            sre[d] = (d == 0) ? 1.0f : 0.0f;
            sim_[d] = 0.0f;
        }
    } else {
        const float inv = 1.0f / sn;
        for (int d = tid; d < DIM; d += 256) {
            sre[d] *= inv; sim_[d] *= inv;
        }
    }
    __syncthreads();

    run_ansatz(sre, sim_, ch, sh, tid);

    for (int w = 0; w < NQ; ++w) {
        const int bp = NQ - 1 - w, m = 1 << bp;
        float xr = 0.0f, xi = 0.0f, zz = 0.0f;
        for (int p = tid; p < PAIR; p += 256) {
            const int i0 = ((p & ~(m - 1)) << 1) | (p & (m - 1));
            const int i1 = i0 | m;
            const float a0r = sre[i0], a0i = sim_[i0];
            const float a1r = sre[i1], a1i = sim_[i1];
            xr += a0r * a1r + a0i * a1i;   // Re(conj(a0)*a1)
            xi += a0r * a1i - a0i * a1r;   // Im(conj(a0)*a1)
            zz += a0r * a0r + a0i * a0i - a1r * a1r - a1i * a1i;
        }
        red[tid] = xr; __syncthreads();
        for (int s = 128; s > 0; s >>= 1) { if (tid < s) red[tid] += red[tid + s]; __syncthreads(); }
        if (tid == 0) out[w] = 2.0f * red[0];
        __syncthreads();
        red[tid] = xi; __syncthreads();
        for (int s = 128; s > 0; s >>= 1) { if (tid < s) red[tid] += red[tid + s]; __syncthreads(); }
        if (tid == 0) out[NQ + w] = 2.0f * red[0];
        __syncthreads();
        red[tid] = zz; __syncthreads();
        for (int s = 128; s > 0; s >>= 1) { if (tid < s) red[tid] += red[tid + s]; __syncthreads(); }
        if (tid == 0) out[2 * NQ + w] = red[0];
        __syncthreads();
    }
}

// ---------------------------------------------------------------------------
extern "C" void kernel_launch(void* const* d_in, const int* in_sizes, int n_in,
                              void* d_out, int out_size, void* d_ws, size_t ws_size,
                              hipStream_t stream) {
    const float* tokens_emb  = (const float*)d_in[0];  // [1024*512]
    const float* W_ang       = (const float*)d_in[1];  // [48*512]
    const float* b_ang       = (const float*)d_in[2];  // [48]
    const float* qsvt_coeffs = (const float*)d_in[3];  // [4]
    const float* lcu_re      = (const float*)d_in[4];  // [1024]
    const float* lcu_im      = (const float*)d_in[5];  // [1024]
    const float* ff_params   = (const float*)d_in[6];  // [48]
    float* out = (float*)d_out;                        // [36]

    // workspace carve-up (needs ~33.9 MB)
    float* ws      = (float*)d_ws;
    float* angles  = ws;                               // 1024*48
    float* wre     = angles + (size_t)LTOK * NPAR;     // 1024
    float* wim     = wre + LTOK;                       // 1024
    float* mono    = wim + LTOK;                       // 2*4096
    float* acc     = mono + 2 * DIM;                   // 2*4096
    float* evolved = acc + 2 * DIM;                    // 1024 * 2*4096

    angles_wmma_kernel<<<192, 32, 0, stream>>>(tokens_emb, W_ang, b_ang, angles);
    prep_kernel<<<1, 1024, 0, stream>>>(lcu_re, lcu_im, qsvt_coeffs, wre, wim, mono, acc);

    for (int k = 1; k <= 3; ++k) {
        norm_kernel<<<1, 256, 0, stream>>>(mono);
        evolve_kernel<<<LTOK, 256, 0, stream>>>(mono, angles, wre, wim, evolved);
        reduce_kernel<<<DIM / 256, 256, 0, stream>>>(evolved, qsvt_coeffs, k, mono, acc);
    }

    final_kernel<<<1, 256, 0, stream>>>(acc, qsvt_coeffs, ff_params, out);
}